// GCN_25314537243163
// MI455X (gfx1250) — compile-verified
//
#include <hip/hip_runtime.h>

typedef float v2f __attribute__((ext_vector_type(2)));
typedef float v8f __attribute__((ext_vector_type(8)));

#define IN_F  512
#define HID_F 256
#define OUT_F 128

#define WMMA_F32(a, b, c) \
  __builtin_amdgcn_wmma_f32_16x16x4_f32(false, (a), false, (b), (short)0, (c), false, false)

// ---------------------------------------------------------------- zeroing
__global__ void zero_f4(float4* __restrict__ p, long long n4) {
  long long i = (long long)blockIdx.x * blockDim.x + threadIdx.x;
  if (i < n4) p[i] = make_float4(0.f, 0.f, 0.f, 0.f);
}

// ---------------------------------------------------------------- degree
__global__ void degree_k(const int* __restrict__ dst, float* __restrict__ deg, int E) {
  int i = blockIdx.x * blockDim.x + threadIdx.x;
  if (i < E) atomicAdd(&deg[dst[i]], 1.0f);
}

__global__ void invdeg_k(float* __restrict__ deg, int n) {
  int i = blockIdx.x * blockDim.x + threadIdx.x;
  if (i < n) deg[i] = 1.0f / fmaxf(deg[i], 1.0f);
}

// ---------------------------------------------------------------- WMMA GEMM
// C[M,N] = A[M,K] @ B[K,N], fp32 row-major, V_WMMA_F32_16X16X4_F32.
// One wave computes a 32(M) x 64(N) strip: two 16-row tiles share the same
// four B fragments -> 8 WMMAs per 10 loads per k-step.
// Block = 128 threads = 4 waves stacked in M (block tile 128x64).
// Requires M%16==0, N%64==0, K%4==0.
__global__ __launch_bounds__(128)
void gemm_wmma_f32(const float* __restrict__ A, const float* __restrict__ B,
                   float* __restrict__ C, int M, int N, int K) {
  const int lane = threadIdx.x & 31;
  const int wave = threadIdx.x >> 5;
  const int m0 = (blockIdx.x * 4 + wave) * 32;
  const int n0 = blockIdx.y * 64;
  if (m0 >= M) return;                  // wave-uniform: EXEC stays all-1s
  const bool has2 = (m0 + 16) < M;      // wave-uniform

  const int half = lane >> 4;           // 0: lanes 0-15, 1: lanes 16-31
  const int l15  = lane & 15;

  // A 16x4 layout: lanes 0-15 row=l15 hold K=k,k+1 ; lanes 16-31 hold K=k+2,k+3
  const int row0 = m0 + l15;
  int row1 = m0 + 16 + l15;
  if (row1 >= M) row1 = M - 1;          // clamp OOB reads; results discarded
  const float* Ap0 = A + (size_t)row0 * K + half * 2;
  const float* Ap1 = A + (size_t)row1 * K + half * 2;
  // B 4x16 layout: VGPR0 = row k+2*half, VGPR1 = row k+1+2*half, col = n0+l15
  const float* Bp = B + (size_t)(half * 2) * N + n0 + l15;

  v8f c00 = {}, c01 = {}, c02 = {}, c03 = {};
  v8f c10 = {}, c11 = {}, c12 = {}, c13 = {};

  for (int k = 0; k < K; k += 4) {
    v2f a0 = *(const v2f*)(Ap0 + k);
    v2f a1 = *(const v2f*)(Ap1 + k);
    const float* bk = Bp + (size_t)k * N;
    v2f b0, b1, b2, b3;
    b0.x = bk[0];   b0.y = bk[(size_t)N + 0];
    b1.x = bk[16];  b1.y = bk[(size_t)N + 16];
    b2.x = bk[32];  b2.y = bk[(size_t)N + 32];
    b3.x = bk[48];  b3.y = bk[(size_t)N + 48];
    c00 = WMMA_F32(a0, b0, c00);
    c01 = WMMA_F32(a0, b1, c01);
    c02 = WMMA_F32(a0, b2, c02);
    c03 = WMMA_F32(a0, b3, c03);
    c10 = WMMA_F32(a1, b0, c10);
    c11 = WMMA_F32(a1, b1, c11);
    c12 = WMMA_F32(a1, b2, c12);
    c13 = WMMA_F32(a1, b3, c13);
  }

  // C/D layout: VGPR j -> row m0 + j + 8*half, col n0 + l15 (+16/32/48)
  float* Cp0 = C + (size_t)(m0 + half * 8) * N + n0 + l15;
#pragma unroll
  for (int j = 0; j < 8; ++j) {
    float* r = Cp0 + (size_t)j * N;
    r[0]  = c00[j];
    r[16] = c01[j];
    r[32] = c02[j];
    r[48] = c03[j];
  }
  if (has2) {
    float* Cp1 = C + (size_t)(m0 + 16 + half * 8) * N + n0 + l15;
#pragma unroll
    for (int j = 0; j < 8; ++j) {
      float* r = Cp1 + (size_t)j * N;
      r[0]  = c10[j];
      r[16] = c11[j];
      r[32] = c12[j];
      r[48] = c13[j];
    }
  }
}

// ---------------------------------------------------------------- edge scatter
// For each (edge, 4-feature chunk): agg[dst] += w * hw[src]. Coalesced float4
// gather, 4x global_atomic_add_f32 scatter. fshift = log2(F/4).
__global__ void edge_scatter(const float* __restrict__ hw, const float* __restrict__ ew,
                             const int* __restrict__ src, const int* __restrict__ dst,
                             float* __restrict__ agg, long long total, int fshift, int F) {
  long long idx = (long long)blockIdx.x * blockDim.x + threadIdx.x;
  if (idx >= total) return;
  int e  = (int)(idx >> fshift);
  int f4 = (int)(idx & ((1 << fshift) - 1)) << 2;
  float w = ew[e];
  const float4 v = *(const float4*)(hw + (size_t)src[e] * F + f4);
  float* p = agg + (size_t)dst[e] * F + f4;
  atomicAdd(p + 0, v.x * w);
  atomicAdd(p + 1, v.y * w);
  atomicAdd(p + 2, v.z * w);
  atomicAdd(p + 3, v.w * w);
}

// ---------------------------------------------------------------- norm + bias (+relu)
__global__ void bias_norm_act(float* __restrict__ data, const float* __restrict__ inv,
                              const float* __restrict__ bias, long long n,
                              int fshift, int do_relu) {
  long long i = (long long)blockIdx.x * blockDim.x + threadIdx.x;
  if (i >= n) return;
  int row = (int)(i >> fshift);
  int col = (int)(i & ((1 << fshift) - 1));
  float v = data[i] * inv[row] + bias[col];
  data[i] = do_relu ? fmaxf(v, 0.f) : v;
}

// ---------------------------------------------------------------- launcher
extern "C" void kernel_launch(void* const* d_in, const int* in_sizes, int n_in,
                              void* d_out, int out_size, void* d_ws, size_t ws_size,
                              hipStream_t stream) {
  const float* x    = (const float*)d_in[0];
  const float* W1   = (const float*)d_in[1];
  const float* b1   = (const float*)d_in[2];
  const float* W2   = (const float*)d_in[3];
  const float* b2   = (const float*)d_in[4];
  const float* ew   = (const float*)d_in[5];
  const int*   esrc = (const int*)d_in[6];
  const int*   edst = (const int*)d_in[7];
  float* out = (float*)d_out;

  const int M = in_sizes[0] / IN_F;   // 50000 nodes
  const int E = in_sizes[5];          // 850000 edges

  // workspace layout: inv[Mpad] | bufA[M*HID_F] | bufB[M*HID_F]
  float* inv  = (float*)d_ws;
  const long long Mpad = ((long long)M + 63) & ~63LL;
  float* bufA = inv + Mpad;
  float* bufB = bufA + (long long)M * HID_F;

  const int T = 256;

  // 1) zero deg, agg1, out
  {
    long long n4 = (long long)M / 4;
    zero_f4<<<dim3((unsigned)((n4 + T - 1) / T)), dim3(T), 0, stream>>>((float4*)inv, n4);
    long long nB4 = (long long)M * HID_F / 4;
    zero_f4<<<dim3((unsigned)((nB4 + T - 1) / T)), dim3(T), 0, stream>>>((float4*)bufB, nB4);
    long long nO4 = (long long)M * OUT_F / 4;
    zero_f4<<<dim3((unsigned)((nO4 + T - 1) / T)), dim3(T), 0, stream>>>((float4*)out, nO4);
  }

  // 2) structural in-degree, then inv = 1/max(deg,1) in place
  degree_k<<<dim3((E + T - 1) / T), dim3(T), 0, stream>>>(edst, inv, E);
  invdeg_k<<<dim3((M + T - 1) / T), dim3(T), 0, stream>>>(inv, M);

  // 3) hw1 = x @ W1  [M, HID_F]
  {
    dim3 g((M + 127) / 128, HID_F / 64);
    gemm_wmma_f32<<<g, dim3(128), 0, stream>>>(x, W1, bufA, M, HID_F, IN_F);
  }

  // 4) agg1 += w * hw1[src]
  {
    long long total = (long long)E * (HID_F / 4);
    edge_scatter<<<dim3((unsigned)((total + T - 1) / T)), dim3(T), 0, stream>>>(
        bufA, ew, esrc, edst, bufB, total, 6 /*log2(64)*/, HID_F);
  }

  // 5) h = relu(agg1 * inv + b1) in place
  {
    long long n = (long long)M * HID_F;
    bias_norm_act<<<dim3((unsigned)((n + T - 1) / T)), dim3(T), 0, stream>>>(
        bufB, inv, b1, n, 8 /*log2(256)*/, 1);
  }

  // 6) hw2 = h @ W2  [M, OUT_F]
  {
    dim3 g((M + 127) / 128, OUT_F / 64);
    gemm_wmma_f32<<<g, dim3(128), 0, stream>>>(bufB, W2, bufA, M, OUT_F, HID_F);
  }

  // 7) out += w * hw2[src]
  {
    long long total = (long long)E * (OUT_F / 4);
    edge_scatter<<<dim3((unsigned)((total + T - 1) / T)), dim3(T), 0, stream>>>(
        bufA, ew, esrc, edst, out, total, 5 /*log2(32)*/, OUT_F);
  }

  // 8) out = out * inv + b2 in place
  {
    long long n = (long long)M * OUT_F;
    bias_norm_act<<<dim3((unsigned)((n + T - 1) / T)), dim3(T), 0, stream>>>(
        out, inv, b2, n, 7 /*log2(128)*/, 0);
  }
}